// SlotMemoryRecurrentLM_34368328303154
// MI455X (gfx1250) — compile-verified
//
#include <hip/hip_runtime.h>
#include <hip/hip_bf16.h>
#include <math.h>

typedef float v2f __attribute__((ext_vector_type(2)));
typedef float v8f __attribute__((ext_vector_type(8)));

__device__ __forceinline__ float sigmoidf_(float x) { return 1.0f / (1.0f + expf(-x)); }

#define GRU_WGS 64

// Async 16B global->LDS copy (per-lane), tracked by ASYNCcnt.
__device__ __forceinline__ void async_copy16(unsigned lds_addr, const float* gsrc) {
    asm volatile("global_load_async_to_lds_b128 %0, %1, off"
                 :: "v"(lds_addr), "v"(gsrc) : "memory");
}
__device__ __forceinline__ void wait_async0() {
    asm volatile("s_wait_asynccnt 0" ::: "memory");
}
__device__ __forceinline__ unsigned lds_addr_of(const void* p) {
    return (unsigned)(unsigned long long)p;   // flat LDS ptr: addr[31:0] = LDS offset
}

// ---------------------------------------------------------------------------
// Generic f32 WMMA GEMM:  C[M,N] = act( A[M,K] * Bw[N,K]^T + bias[N] )
// Block = 256 threads = 8 waves (2x4); each wave owns a 32x32 C tile
// (2x2 wmma sub-tiles, A/B fragments reused pairwise) -> 64(M) x 128(N)
// macro-tile. K staged in LDS in chunks of 32 via double-buffered
// GLOBAL_LOAD_ASYNC_TO_LDS_B128 (copy of stage i+1 overlaps wmma of stage i).
// M, K must be multiples of 64/32 (true for all call sites); N is guarded.
// act: 0=none, 1=sigmoid, 2=tanh, 3=relu^2
// ---------------------------------------------------------------------------
__global__ __launch_bounds__(256) void gemm_f32_wmma(
    const float* __restrict__ A, const float* __restrict__ Bw,
    const float* __restrict__ bias, float* __restrict__ C,
    const int* __restrict__ gather, int M, int N, int K, int act)
{
    __shared__ float As[2][64][36];    // 36-float pitch: 16B aligned, conflict-free
    __shared__ float Bs[2][128][36];

    const int tid  = threadIdx.x;
    const int wave = tid >> 5;
    const int lane = tid & 31;
    const int wr   = wave >> 2;      // 0..1  (M 32-tile)
    const int wc   = wave & 3;       // 0..3  (N 32-tile)
    const int half = lane >> 4;      // 0/1
    const int l16  = lane & 15;

    // A loader: 64 rows x 32 cols, 2x16B per thread
    const int ar = tid >> 2;
    const int ac = (tid & 3) * 8;
    // B loader: 128 rows x 32 cols, 4x16B per thread
    const int brw = tid >> 1;
    const int bc  = (tid & 1) * 16;

    const int  am    = blockIdx.y * 64 + ar;
    const long abase = (long)(gather ? gather[am] : am) * (long)K;
    int bn = blockIdx.x * 128 + brw;
    if (bn > N - 1) bn = N - 1;                 // clamp (keeps loads in-bounds)
    const long bbase = (long)bn * (long)K;

    const float* Ap = A  + abase;
    const float* Bp = Bw + bbase;

    auto issueStage = [&](int k0, int buf) {
        async_copy16(lds_addr_of(&As[buf][ar][ac]),       Ap + k0 + ac);
        async_copy16(lds_addr_of(&As[buf][ar][ac + 4]),   Ap + k0 + ac + 4);
        async_copy16(lds_addr_of(&Bs[buf][brw][bc]),      Bp + k0 + bc);
        async_copy16(lds_addr_of(&Bs[buf][brw][bc + 4]),  Bp + k0 + bc + 4);
        async_copy16(lds_addr_of(&Bs[buf][brw][bc + 8]),  Bp + k0 + bc + 8);
        async_copy16(lds_addr_of(&Bs[buf][brw][bc + 12]), Bp + k0 + bc + 12);
    };

    v8f acc[2][2] = {};

    issueStage(0, 0);
    int buf = 0;
    for (int k0 = 0; k0 < K; k0 += 32, buf ^= 1) {
        wait_async0();        // my lanes' copies into As/Bs[buf] complete
        __syncthreads();      // => all waves' copies complete; prev reads of buf^1 done
        if (k0 + 32 < K) issueStage(k0 + 32, buf ^ 1);   // overlap with wmma below
#pragma unroll
        for (int kk = 0; kk < 32; kk += 4) {
            const int kf = kk + 2 * half;       // v0 -> K=kf, v1 -> K=kf+1
            v2f fa0, fa1, fb0, fb1;
            fa0[0] = As[buf][wr * 32 + l16][kf];      fa0[1] = As[buf][wr * 32 + l16][kf + 1];
            fa1[0] = As[buf][wr * 32 + 16 + l16][kf]; fa1[1] = As[buf][wr * 32 + 16 + l16][kf + 1];
            fb0[0] = Bs[buf][wc * 32 + l16][kf];      fb0[1] = Bs[buf][wc * 32 + l16][kf + 1];
            fb1[0] = Bs[buf][wc * 32 + 16 + l16][kf]; fb1[1] = Bs[buf][wc * 32 + 16 + l16][kf + 1];
            acc[0][0] = __builtin_amdgcn_wmma_f32_16x16x4_f32(false, fa0, false, fb0, (short)0, acc[0][0], false, false);
            acc[0][1] = __builtin_amdgcn_wmma_f32_16x16x4_f32(false, fa0, false, fb1, (short)0, acc[0][1], false, false);
            acc[1][0] = __builtin_amdgcn_wmma_f32_16x16x4_f32(false, fa1, false, fb0, (short)0, acc[1][0], false, false);
            acc[1][1] = __builtin_amdgcn_wmma_f32_16x16x4_f32(false, fa1, false, fb1, (short)0, acc[1][1], false, false);
        }
    }

    const int rbase = blockIdx.y * 64 + wr * 32;
    const int cbase = blockIdx.x * 128 + wc * 32;
#pragma unroll
    for (int ni = 0; ni < 2; ++ni) {
        const int col = cbase + ni * 16 + l16;
        if (col >= N) continue;
        const float bv = bias ? bias[col] : 0.0f;
#pragma unroll
        for (int mi = 0; mi < 2; ++mi) {
            const int row0 = rbase + mi * 16 + 8 * half;
#pragma unroll
            for (int i = 0; i < 8; ++i) {
                float v = acc[mi][ni][i] + bv;
                if (act == 1)      v = sigmoidf_(v);
                else if (act == 2) v = tanhf(v);
                else if (act == 3) { v = v > 0.0f ? v : 0.0f; v = v * v; }
                C[(size_t)(row0 + i) * (size_t)N + (size_t)col] = v;
            }
        }
    }
}

// ---------------------------------------------------------------------------
// GRU scan, 64 workgroups. WG g owns hidden units [g*16, g*16+16) i.e. weight
// rows {gate*1024 + u} for its 16 units x 3 gates = 48 rows. Batch copies of
// a row sit in adjacent lanes so their weight loads coalesce. h[4][1024]
// ping-pongs through global memory; steps separated by a device-scope
// release/acquire spin barrier (all 64 WGs trivially co-resident).
// ---------------------------------------------------------------------------
__global__ __launch_bounds__(256) void gru_scan_mw(
    const float* __restrict__ gi, const float* __restrict__ w_hh,
    const float* __restrict__ b_hh, float* __restrict__ states,
    float* __restrict__ hbuf, unsigned* __restrict__ counter)
{
    const int t  = threadIdx.x;
    const int u0 = blockIdx.x * 16;
    __shared__ __align__(16) float hls[4 * 1032];   // pad 1032: conflict-free
    __shared__ float dots[48][4];

    const int lcl  = (t < 192) ? (t >> 2) : 0;  // 0..47 (gate*16 + unit)
    const int b4   = t & 3;                     // batch
    const int gate = lcl >> 4;                  // 0..2
    const int uu   = lcl & 15;
    const int row  = gate * 1024 + u0 + uu;
    const float* wrow = w_hh + (size_t)row * 1024;
    const float brow  = b_hh[row];

    for (int step = 0; step < 1024; ++step) {
        const int cur = step & 1, nxt = cur ^ 1;
        {   // stage h_{t-1} into LDS (coalesced, 16 floats/thread)
            const float* src = hbuf + cur * 4096 + t * 16;
            float* dst = hls + ((t * 16) >> 10) * 1032 + ((t * 16) & 1023);
            float4 v0 = *(const float4*)(src);
            float4 v1 = *(const float4*)(src + 4);
            float4 v2 = *(const float4*)(src + 8);
            float4 v3 = *(const float4*)(src + 12);
            *(float4*)(dst) = v0; *(float4*)(dst + 4) = v1;
            *(float4*)(dst + 8) = v2; *(float4*)(dst + 12) = v3;
        }
        __syncthreads();
        if (t < 192) {
            const float* hb = hls + b4 * 1032;
            float acc = brow;
#pragma unroll 4
            for (int k = 0; k < 1024; k += 8) {
                float4 w0 = *(const float4*)(wrow + k);
                float4 w1 = *(const float4*)(wrow + k + 4);
                float4 h0 = *(const float4*)(hb + k);
                float4 h1 = *(const float4*)(hb + k + 4);
                acc = fmaf(w0.x, h0.x, acc); acc = fmaf(w0.y, h0.y, acc);
                acc = fmaf(w0.z, h0.z, acc); acc = fmaf(w0.w, h0.w, acc);
                acc = fmaf(w1.x, h1.x, acc); acc = fmaf(w1.y, h1.y, acc);
                acc = fmaf(w1.z, h1.z, acc); acc = fmaf(w1.w, h1.w, acc);
            }
            dots[lcl][b4] = acc;
        }
        __syncthreads();
        if (t < 64) {                               // gate math for 16 units x 4 batches
            const int u = t >> 2, b = t & 3;
            const size_t girow = ((size_t)b * 1024 + step) * 3072 + u0 + u;
            const float r = sigmoidf_(gi[girow]        + dots[u][b]);
            const float z = sigmoidf_(gi[girow + 1024] + dots[16 + u][b]);
            const float n = tanhf   (gi[girow + 2048]  + r * dots[32 + u][b]);
            const float hp = hls[b * 1032 + u0 + u];
            const float hn = (1.0f - z) * n + z * hp;
            states[((size_t)b * 1024 + step) * 1024 + u0 + u] = hn;
            hbuf[nxt * 4096 + b * 1024 + u0 + u] = hn;
        }
        __syncthreads();
        if (t == 0) {                               // device-scope step barrier
            __hip_atomic_fetch_add(counter, 1u, __ATOMIC_RELEASE, __HIP_MEMORY_SCOPE_AGENT);
            const unsigned target = (unsigned)GRU_WGS * (unsigned)(step + 1);
            while (__hip_atomic_load(counter, __ATOMIC_ACQUIRE, __HIP_MEMORY_SCOPE_AGENT) < target)
                __builtin_amdgcn_s_sleep(1);
        }
        __syncthreads();
    }
}

__global__ void gru_init(float* hbuf, unsigned* counter) {
    const int i = blockIdx.x * blockDim.x + threadIdx.x;
    if (i < 8192) hbuf[i] = 0.0f;
    if (i == 0) *counter = 0u;
}

// ---------------------------------------------------------------------------
// Slot-memory scan: one 256-thread workgroup per batch row.
// Slots [64,128] live in LDS for the whole scan. Emits read_vecs [B,T,128].
// ---------------------------------------------------------------------------
__global__ __launch_bounds__(256) void slot_scan(
    const float* __restrict__ slot_init,
    const float* __restrict__ rq, const float* __restrict__ wq,
    const float* __restrict__ wv, const float* __restrict__ wg,
    float* __restrict__ read_vecs)
{
    const int b = blockIdx.x;
    const int t = threadIdx.x;
    __shared__ float sl[64][128];
    __shared__ float partR[64][4], partW[64][4];
    __shared__ float attnR[64], wsv[64];
    __shared__ float red[2];

    for (int i = t; i < 64 * 128; i += 256)
        sl[i >> 7][i & 127] = slot_init[i];
    __syncthreads();

    const int s = t >> 2, q = t & 3;
    for (int step = 0; step < 1024; ++step) {
        const size_t r128 = ((size_t)b * 1024 + step) * 128;
        const float* rqt = rq + r128;
        const float* wqt = wq + r128;
        const float* wvt = wv + r128;
        const float* wgt = wg + ((size_t)b * 1024 + step) * 64;

        float pr = 0.0f, pw = 0.0f;
        for (int d = q * 32; d < q * 32 + 32; ++d) {
            const float sv = sl[s][d];
            pr = fmaf(rqt[d], sv, pr);
            pw = fmaf(wqt[d], sv, pw);
        }
        partR[s][q] = pr; partW[s][q] = pw;
        __syncthreads();

        if (t < 64) {
            attnR[t] = partR[t][0] + partR[t][1] + partR[t][2] + partR[t][3];
            wsv[t]   = partW[t][0] + partW[t][1] + partW[t][2] + partW[t][3];
        }
        __syncthreads();
        if (t == 0) {
            float mR = -1e30f, mW = -1e30f;
            for (int i = 0; i < 64; ++i) {
                mR = fmaxf(mR, attnR[i]); mW = fmaxf(mW, wsv[i]);
            }
            red[0] = mR; red[1] = mW;
        }
        __syncthreads();
        if (t < 64) {
            attnR[t] = expf(attnR[t] - red[0]);
            wsv[t]   = expf(wsv[t]   - red[1]);
        }
        __syncthreads();
        if (t == 0) {
            float sR = 0.0f, sW = 0.0f;
            for (int i = 0; i < 64; ++i) { sR += attnR[i]; sW += wsv[i]; }
            red[0] = 1.0f / sR; red[1] = 1.0f / sW;
        }
        __syncthreads();
        if (t < 64) {
            attnR[t] *= red[0];
            wsv[t]    = wsv[t] * red[1] * wgt[t];   // write_attn * gate
        }
        __syncthreads();

        if (t < 128) {                              // read vector (pre-update slots)
            float rv = 0.0f;
            for (int ss = 0; ss < 64; ++ss) rv = fmaf(attnR[ss], sl[ss][t], rv);
            read_vecs[r128 + t] = rv;
        }
        __syncthreads();

        for (int i = t; i < 64 * 128; i += 256) {   // slot update
            const int ss = i >> 7, dd = i & 127;
            const float w = wsv[ss];
            sl[ss][dd] = sl[ss][dd] * (1.0f - w) + wvt[dd] * w;
        }
        __syncthreads();
    }
}

// mixed = base + gate * retrieved
__global__ void mix_kernel(const float* __restrict__ base,
                           const float* __restrict__ gate,
                           const float* __restrict__ retr,
                           float* __restrict__ mixed, int n)
{
    const int i = blockIdx.x * blockDim.x + threadIdx.x;
    if (i < n) mixed[i] = fmaf(gate[i], retr[i], base[i]);
}

// ---------------------------------------------------------------------------
extern "C" void kernel_launch(void* const* d_in, const int* in_sizes, int n_in,
                              void* d_out, int out_size, void* d_ws, size_t ws_size,
                              hipStream_t stream) {
    (void)in_sizes; (void)n_in; (void)out_size; (void)ws_size;
    const int*   ids        = (const int*)  d_in[0];
    const float* emb        = (const float*)d_in[1];
    const float* w_ih       = (const float*)d_in[2];
    const float* w_hh       = (const float*)d_in[3];
    const float* b_ih       = (const float*)d_in[4];
    const float* b_hh       = (const float*)d_in[5];
    const float* head_fc_w  = (const float*)d_in[6];
    const float* head_fc_b  = (const float*)d_in[7];
    const float* head_pj_w  = (const float*)d_in[8];
    const float* head_pj_b  = (const float*)d_in[9];
    const float* out_bias   = (const float*)d_in[10];
    const float* slot_init  = (const float*)d_in[11];
    const float* rq_w = (const float*)d_in[12]; const float* rq_b = (const float*)d_in[13];
    const float* wq_w = (const float*)d_in[14]; const float* wq_b = (const float*)d_in[15];
    const float* wv_w = (const float*)d_in[16]; const float* wv_b = (const float*)d_in[17];
    const float* wg_w = (const float*)d_in[18]; const float* wg_b = (const float*)d_in[19];
    const float* rm_w = (const float*)d_in[20]; const float* rm_b = (const float*)d_in[21];
    const float* s2f_w = (const float*)d_in[22]; const float* s2f_b = (const float*)d_in[23];
    float* out = (float*)d_out;

    const int BT = 4096, E = 512, H = 1024, V = 32000;

    char* ws = (char*)d_ws;
    float* gi        = (float*)(ws);                         // 4096x3072
    float* states    = (float*)(ws + 50331648);              // 4096x1024
    float* rqv       = (float*)(ws + 67108864);              // 4096x128
    float* wqv       = (float*)(ws + 69206016);              // 4096x128
    float* wvv       = (float*)(ws + 71303168);              // 4096x128
    float* wgv       = (float*)(ws + 73400320);              // 4096x64
    float* read_vecs = (float*)(ws + 74448896);              // 4096x128
    float* head      = (float*)(ws + 76546048);              // 4096x2048
    float* base      = (float*)(ws + 110100480);             // 4096x512
    float* gate      = (float*)(ws + 118489088);             // 4096x512
    float* retr      = (float*)(ws + 126877696);             // 4096x512
    float* mixed     = (float*)(ws + 135266304);             // 4096x512
    float* hbuf      = (float*)(ws + 143654912);             // 2x4x1024
    unsigned* counter= (unsigned*)(ws + 143687680);

    // 1. gi = emb[ids] @ w_ih^T + b_ih   [4096,3072], K=512, gathered A
    gemm_f32_wmma<<<dim3(3072 / 128, BT / 64), 256, 0, stream>>>(
        emb, w_ih, b_ih, gi, ids, BT, 3072, E, 0);
    // 2. GRU scan -> states [4096,1024]
    gru_init<<<32, 256, 0, stream>>>(hbuf, counter);
    gru_scan_mw<<<GRU_WGS, 256, 0, stream>>>(gi, w_hh, b_hh, states, hbuf, counter);
    // 3. slot projections
    gemm_f32_wmma<<<dim3(1, BT / 64), 256, 0, stream>>>(states, rq_w, rq_b, rqv, nullptr, BT, 128, H, 0);
    gemm_f32_wmma<<<dim3(1, BT / 64), 256, 0, stream>>>(states, wq_w, wq_b, wqv, nullptr, BT, 128, H, 0);
    gemm_f32_wmma<<<dim3(1, BT / 64), 256, 0, stream>>>(states, wv_w, wv_b, wvv, nullptr, BT, 128, H, 2);
    gemm_f32_wmma<<<dim3(1, BT / 64), 256, 0, stream>>>(states, wg_w, wg_b, wgv, nullptr, BT, 64,  H, 1);
    // 4. slot memory scan -> read_vecs [4096,128]
    slot_scan<<<4, 256, 0, stream>>>(slot_init, rqv, wqv, wvv, wgv, read_vecs);
    // 5. retrieved = read_vecs @ s2f^T  [4096,512], K=128
    gemm_f32_wmma<<<dim3(4, BT / 64), 256, 0, stream>>>(read_vecs, s2f_w, s2f_b, retr, nullptr, BT, 512, 128, 0);
    // 6. head = relu(states @ head_fc^T)^2  [4096,2048]
    gemm_f32_wmma<<<dim3(16, BT / 64), 256, 0, stream>>>(states, head_fc_w, head_fc_b, head, nullptr, BT, 2048, H, 3);
    // 7. base = head @ head_proj^T  [4096,512], K=2048
    gemm_f32_wmma<<<dim3(4, BT / 64), 256, 0, stream>>>(head, head_pj_w, head_pj_b, base, nullptr, BT, 512, 2048, 0);
    // 8. gate = sigmoid(states @ rm^T)  [4096,512]
    gemm_f32_wmma<<<dim3(4, BT / 64), 256, 0, stream>>>(states, rm_w, rm_b, gate, nullptr, BT, 512, H, 1);
    // 9. mixed = base + gate * retrieved
    mix_kernel<<<(BT * E + 255) / 256, 256, 0, stream>>>(base, gate, retr, mixed, BT * E);
    // 10. logits = mixed @ emb^T + out_bias  [4096,32000], K=512
    gemm_f32_wmma<<<dim3(V / 128, BT / 64), 256, 0, stream>>>(mixed, emb, out_bias, out, nullptr, BT, V, E, 0);
}